// ParametricExtendedKalmanFilter_12455405159153
// MI455X (gfx1250) — compile-verified
//
#include <hip/hip_runtime.h>

// ----------------------------------------------------------------------------
// Parametric EKF on MI455X (gfx1250): single persistent workgroup, wave32,
// f32 WMMA (V_WMMA_F32_16X16X4_F32) for every GEMM, LDS-resident state,
// in-place pivot-free Gauss-Jordan inverse of the SPD innovation matrix.
// ----------------------------------------------------------------------------

#define DIM_X   128
#define DIM_Z   64
#define TSTEPS  2048
#define NTHREADS 256   // 8 wave32s on one WGP

typedef float v2f __attribute__((ext_vector_type(2)));
typedef float v8f __attribute__((ext_vector_type(8)));

// LDS layout (float offsets)
#define OFF_P    0        // 128x128 covariance            (64 KB)
#define OFF_F    16384    // F / (I - K H) factor          (64 KB)
#define OFF_T1   32768    // scratch: B1 / B2 / F@P / fac@P (64 KB)
#define OFF_HT   49152    // H^T  128x64                   (32 KB)
#define OFF_PHT  57344    // P@H^T, later K@R              (32 KB)
#define OFF_S    65536    // S -> S^-1 in place 64x64      (16 KB)
#define OFF_K    69632    // Kalman gain 128x64            (32 KB)
#define OFF_VEC  77824    // xp,xf,tv,sv (128 ea), y,colk (64 ea)
#define SMEM_FLOATS (OFF_VEC + 4*128 + 2*64)
#define SMEM_BYTES  (SMEM_FLOATS * sizeof(float))   // 313856 B < 320 KB

// ----------------------------------------------------------------------------
// Tiled C[MxN] (+)= A[MxK] * B[KxN] using V_WMMA_F32_16X16X4_F32.
// One 16x16 tile per wave per pass; K swept in chunks of 4.
// Fragment layouts per CDNA5 ISA 7.12.2:
//   A 16x4 : lanes 0-15 row M, VGPR r -> K=k0+r ; lanes 16-31 -> K=k0+2+r
//   B 4x16 : lanes 0-15 col N, VGPR r -> K=k0+r ; lanes 16-31 -> K=k0+2+r
//   C 16x16: VGPR r -> row r (lanes 0-15) / row r+8 (lanes 16-31)
// TA/TB: logical transpose via swizzled fragment indexing.
// ----------------------------------------------------------------------------
template <bool TA, bool TB, bool ACC>
__device__ inline void mm16(float* __restrict__ C, int ldc,
                            const float* __restrict__ A, int lda,
                            const float* __restrict__ B, int ldb,
                            int M, int N, int Kd)
{
    const int lane = threadIdx.x & 31;
    const int wave = threadIdx.x >> 5;
    const int nw   = NTHREADS >> 5;
    const int half = lane >> 4;       // 0: lanes 0-15, 1: lanes 16-31
    const int l16  = lane & 15;
    const int tilesN = N >> 4;
    const int ntiles = (M >> 4) * tilesN;

    for (int t = wave; t < ntiles; t += nw) {
        const int m0 = (t / tilesN) << 4;
        const int n0 = (t % tilesN) << 4;

        v8f acc = {};
        if (ACC) {
#pragma unroll
            for (int r = 0; r < 8; ++r)
                acc[r] = C[(m0 + half * 8 + r) * ldc + (n0 + l16)];
        }

        for (int k0 = 0; k0 < Kd; k0 += 4) {
            v2f af, bf;
            const int am = m0 + l16;
            const int bn = n0 + l16;
#pragma unroll
            for (int r = 0; r < 2; ++r) {
                const int kk = k0 + half * 2 + r;
                af[r] = TA ? A[kk * lda + am] : A[am * lda + kk];
                bf[r] = TB ? B[bn * ldb + kk] : B[kk * ldb + bn];
            }
            acc = __builtin_amdgcn_wmma_f32_16x16x4_f32(
                false, af, false, bf, (short)0, acc, false, false);
        }

#pragma unroll
        for (int r = 0; r < 8; ++r)
            C[(m0 + half * 8 + r) * ldc + (n0 + l16)] = acc[r];
    }
}

__global__ void __launch_bounds__(NTHREADS, 1)
ekf_persistent_kernel(const float* __restrict__ z,   // (T, 64)
                      const float* __restrict__ x0,  // (128,)
                      const float* __restrict__ P0,  // (128,128)
                      const float* __restrict__ Q,   // (128,128)
                      const float* __restrict__ R,   // (64,64)
                      const float* __restrict__ Wf1, // (128,128)
                      const float* __restrict__ Wf2, // (128,128)
                      const float* __restrict__ Wh1, // (128,128)
                      const float* __restrict__ Wh2, // (128,64)
                      float* __restrict__ out)       // (T,128)
{
    extern __shared__ float sm[];
    float* P    = sm + OFF_P;
    float* Fb   = sm + OFF_F;
    float* T1   = sm + OFF_T1;
    float* Ht   = sm + OFF_HT;
    float* PHt  = sm + OFF_PHT;   // later reused as K@R
    float* Sb   = sm + OFF_S;
    float* Kb   = sm + OFF_K;
    float* xp   = sm + OFF_VEC;
    float* xf   = xp + 128;
    float* tv   = xf + 128;
    float* sv   = tv + 128;
    float* yv   = sv + 128;
    float* colk = yv + 64;

    const int tid = threadIdx.x;

    // ---- init state from inputs (fresh every launch: graph-replay safe) ----
    for (int i = tid; i < DIM_X * DIM_X; i += NTHREADS) P[i] = P0[i];
    if (tid < DIM_X) xp[tid] = x0[tid];
    __syncthreads();

    for (int t = 0; t < TSTEPS; ++t) {
        // 1) u = xp@Wf1 ; tv=tanh(u) ; sv=sech^2(u)
        if (tid < DIM_X) {
            float u = 0.f;
            for (int k = 0; k < DIM_X; ++k) u += xp[k] * Wf1[k * DIM_X + tid];
            float th = tanhf(u);
            tv[tid] = th;
            sv[tid] = 1.f - th * th;
        }
        __syncthreads();
        // 2) x_prior = xp + 0.01*(tv@Wf2)   3) B1 = 0.01*diag(sv)@Wf2 -> T1
        if (tid < DIM_X) {
            float a = 0.f;
            for (int k = 0; k < DIM_X; ++k) a += tv[k] * Wf2[k * DIM_X + tid];
            xf[tid] = xp[tid] + 0.01f * a;
        }
        for (int i = tid; i < DIM_X * DIM_X; i += NTHREADS)
            T1[i] = 0.01f * sv[i >> 7] * Wf2[i];
        __syncthreads();
        // 4) F = I + Wf1 @ B1
        mm16<false, false, false>(Fb, DIM_X, Wf1, DIM_X, T1, DIM_X, DIM_X, DIM_X, DIM_X);
        __syncthreads();
        if (tid < DIM_X) Fb[tid * DIM_X + tid] += 1.f;
        __syncthreads();
        // 5) T1 = F @ P
        mm16<false, false, false>(T1, DIM_X, Fb, DIM_X, P, DIM_X, DIM_X, DIM_X, DIM_X);
        __syncthreads();
        // 6) P_prior = T1 @ F^T + Q
        mm16<false, true, false>(P, DIM_X, T1, DIM_X, Fb, DIM_X, DIM_X, DIM_X, DIM_X);
        __syncthreads();
        for (int i = tid; i < DIM_X * DIM_X; i += NTHREADS) P[i] += Q[i];
        // 7) v = xf@Wh1 ; tv=tanh(v) ; sv=sech^2(v)   (tv/sv reuse is safe)
        if (tid < DIM_X) {
            float v = 0.f;
            for (int k = 0; k < DIM_X; ++k) v += xf[k] * Wh1[k * DIM_X + tid];
            float th = tanhf(v);
            tv[tid] = th;
            sv[tid] = 1.f - th * th;
        }
        __syncthreads();
        // 8) innovation y = z_t - tv@Wh2     9) B2 = diag(sv)@Wh2 -> T1[:8192]
        if (tid < DIM_Z) {
            float a = 0.f;
            for (int k = 0; k < DIM_X; ++k) a += tv[k] * Wh2[k * DIM_Z + tid];
            yv[tid] = z[(size_t)t * DIM_Z + tid] - a;
        }
        for (int i = tid; i < DIM_X * DIM_Z; i += NTHREADS)
            T1[i] = sv[i >> 6] * Wh2[i];
        __syncthreads();
        // 10) H^T = Wh1 @ B2   (128x64)
        mm16<false, false, false>(Ht, DIM_Z, Wh1, DIM_X, T1, DIM_Z, DIM_X, DIM_Z, DIM_X);
        __syncthreads();
        // 11) PHt = P_prior @ H^T
        mm16<false, false, false>(PHt, DIM_Z, P, DIM_X, Ht, DIM_Z, DIM_X, DIM_Z, DIM_X);
        __syncthreads();
        // 12) S = H @ PHt + R = Ht^T @ PHt + R
        mm16<true, false, false>(Sb, DIM_Z, Ht, DIM_Z, PHt, DIM_Z, DIM_Z, DIM_Z, DIM_X);
        __syncthreads();
        for (int i = tid; i < DIM_Z * DIM_Z; i += NTHREADS) Sb[i] += R[i];
        __syncthreads();
        // 13) S <- S^-1 in place (pivot-free Gauss-Jordan; S is SPD)
        for (int k = 0; k < DIM_Z; ++k) {
            if (tid < DIM_Z) colk[tid] = Sb[tid * DIM_Z + k];
            __syncthreads();
            const float pinv = 1.f / colk[k];
            if (tid < DIM_Z)
                Sb[k * DIM_Z + tid] = (tid == k) ? pinv : Sb[k * DIM_Z + tid] * pinv;
            __syncthreads();
            for (int idx = tid; idx < DIM_Z * DIM_Z; idx += NTHREADS) {
                const int i = idx >> 6, j = idx & 63;
                if (i != k) {
                    const float f   = colk[i];
                    const float skj = Sb[k * DIM_Z + j];   // already scaled
                    Sb[idx] = (j == k) ? (-f * skj) : (Sb[idx] - f * skj);
                }
            }
            __syncthreads();
        }
        // 14) K = PHt @ S^-1
        mm16<false, false, false>(Kb, DIM_Z, PHt, DIM_Z, Sb, DIM_Z, DIM_X, DIM_Z, DIM_Z);
        __syncthreads();
        // 15) x_post = xf + K@y  -> output + carry    16) KR = K@R -> PHt
        if (tid < DIM_X) {
            float a = 0.f;
            for (int k = 0; k < DIM_Z; ++k) a += Kb[tid * DIM_Z + k] * yv[k];
            const float xn = xf[tid] + a;
            out[(size_t)t * DIM_X + tid] = xn;
            xp[tid] = xn;
        }
        mm16<false, false, false>(PHt, DIM_Z, Kb, DIM_Z, R, DIM_Z, DIM_X, DIM_Z, DIM_Z);
        __syncthreads();
        // 17) factor = I - K @ Ht^T  -> Fb
        mm16<false, true, false>(Fb, DIM_X, Kb, DIM_Z, Ht, DIM_Z, DIM_X, DIM_X, DIM_Z);
        __syncthreads();
        for (int i = tid; i < DIM_X * DIM_X; i += NTHREADS) {
            const int r = i >> 7, c = i & 127;
            Fb[i] = ((r == c) ? 1.f : 0.f) - Fb[i];
        }
        __syncthreads();
        // 18) T1 = factor @ P_prior
        mm16<false, false, false>(T1, DIM_X, Fb, DIM_X, P, DIM_X, DIM_X, DIM_X, DIM_X);
        __syncthreads();
        // 19) P = T1 @ factor^T        20) P += KR @ K^T  (Joseph form)
        mm16<false, true, false>(P, DIM_X, T1, DIM_X, Fb, DIM_X, DIM_X, DIM_X, DIM_X);
        __syncthreads();
        mm16<false, true, true>(P, DIM_X, PHt, DIM_Z, Kb, DIM_Z, DIM_X, DIM_X, DIM_Z);
        __syncthreads();
    }
}

extern "C" void kernel_launch(void* const* d_in, const int* in_sizes, int n_in,
                              void* d_out, int out_size, void* d_ws, size_t ws_size,
                              hipStream_t stream) {
    const float* z   = (const float*)d_in[0];
    const float* x0  = (const float*)d_in[1];
    const float* P0  = (const float*)d_in[2];
    const float* Q   = (const float*)d_in[3];
    const float* R   = (const float*)d_in[4];
    const float* Wf1 = (const float*)d_in[5];
    const float* Wf2 = (const float*)d_in[6];
    const float* Wh1 = (const float*)d_in[7];
    const float* Wh2 = (const float*)d_in[8];
    float* out = (float*)d_out;

    // Opt in to >default dynamic LDS (307 KB of the WGP's 320 KB).
    // Host-side attribute set; idempotent and deterministic.
    (void)hipFuncSetAttribute((const void*)ekf_persistent_kernel,
                              hipFuncAttributeMaxDynamicSharedMemorySize,
                              (int)SMEM_BYTES);

    ekf_persistent_kernel<<<1, NTHREADS, SMEM_BYTES, stream>>>(
        z, x0, P0, Q, R, Wf1, Wf2, Wh1, Wh2, out);
}